// VectorQuantization_54477365182886
// MI455X (gfx1250) — compile-verified
//
#include <hip/hip_runtime.h>

// ---------------------------------------------------------------------------
// VectorQuantization eval forward, fused for MI455X (gfx1250, wave32).
//
// reference:  logits = X(8,2048,1024) @ W(640,1024)^T + b
//             view as (16, 655360), argmax per row  -> 16 winners
//             out(8,2048,512) = zeros except 16 rows of 256 gathered from
//             codebook(640,256).
//
// Never materialize logits. Zero-fill output, then fused f32-WMMA GEMM + packed-u64 atomicMax argmax,
// then scatter 16 codebook rows. V_WMMA_F32_16X16X4_F32
// keeps the argmax fp32-faithful. Global->LDS staging uses CDNA5 async copies
// (GLOBAL_LOAD_ASYNC_TO_LDS_B128 + s_wait_asynccnt) with double buffering so
// the next K-chunk's fetch overlaps the current chunk's WMMAs.
// ---------------------------------------------------------------------------

typedef __attribute__((ext_vector_type(2))) float v2f;
typedef __attribute__((ext_vector_type(8))) float v8f;

#define B_DIM   8
#define T_DIM   2048
#define C_DIM   1024
#define G_DIM   2
#define E_DIM   320
#define D_DIM   256
#define K_OUT   (G_DIM * E_DIM)   // 640
#define HALF_T  1024              // T rows per flat argmax group
#define N_GROUP 16                // B*G flat rows

#define BT      64                // block tile: t rows
#define BK      64                // block tile: k cols
#define KC      32                // K(C)-chunk staged in LDS
#define LDS_STR 36                // padded row stride (16B-aligned, 4*(9r%16))
#define NCHUNK  (C_DIM / KC)      // 32

// Order-preserving float -> uint map (monotone under unsigned compare).
__device__ __forceinline__ unsigned fmap(float x) {
    unsigned u = __float_as_uint(x);
    return (u & 0x80000000u) ? ~u : (u | 0x80000000u);
}

__device__ __forceinline__ unsigned long long shflxor_u64(unsigned long long v, int mask) {
    unsigned lo = (unsigned)(v & 0xffffffffull);
    unsigned hi = (unsigned)(v >> 32);
    lo = __shfl_xor(lo, mask, 32);
    hi = __shfl_xor(hi, mask, 32);
    return ((unsigned long long)hi << 32) | lo;
}

// Async 16B global->LDS copy (ASYNCcnt-tracked). GVS addressing:
// mem = SADDR(64b, sgpr pair) + VADDR(32b byte offset) ; LDS dst byte addr in VGPR.
#define ASYNC_B128(ldsaddr, gvoff, sbase)                               \
    asm volatile("global_load_async_to_lds_b128 %0, %1, %2"             \
                 :: "v"(ldsaddr), "v"(gvoff), "s"(sbase) : "memory")

// ---------------------------------------------------------------------------
// Kernel 1: zero the output, init the 16 argmax key slots.
// ---------------------------------------------------------------------------
__global__ void vq_zero_kernel(float4* __restrict__ out,
                               unsigned long long* __restrict__ keys, long n4) {
    long i = (long)blockIdx.x * blockDim.x + threadIdx.x;
    long stride = (long)gridDim.x * blockDim.x;
    float4 z = {0.f, 0.f, 0.f, 0.f};
    for (; i < n4; i += stride) out[i] = z;
    if (blockIdx.x == 0 && threadIdx.x < N_GROUP) keys[threadIdx.x] = 0ull;
}

// ---------------------------------------------------------------------------
// Kernel 2: fused GEMM (X @ W^T + b) + per-group argmax.
// Block = 256 threads (8 waves as 4t x 2k). Each wave owns a 16x32 logits
// tile (two v8f accumulators, A fragment reused). Block tile = 64t x 64k.
// K loop over C in 32-wide chunks, double-buffered LDS filled by async
// global->LDS b128 copies. 16 WMMAs per chunk per wave, 32 chunks.
// Grid = 16 groups * (1024/64) * (640/64) = 2560 blocks.
// ---------------------------------------------------------------------------
__global__ __launch_bounds__(256) void vq_gemm_argmax_kernel(
    const float* __restrict__ X, const float* __restrict__ W,
    const float* __restrict__ bias, unsigned long long* __restrict__ keys) {

    __shared__ float Xs[2][BT][LDS_STR];
    __shared__ float Ws[2][BK][LDS_STR];
    __shared__ unsigned long long wkeys[8];

    const int group = blockIdx.x / 160;     // 0..15  (= b*2 + h)
    const int rem   = blockIdx.x % 160;
    const int tBlk  = rem / 10;             // 0..15  (64 rows each)
    const int kBlk  = rem % 10;             // 0..9   (64 cols each)
    const int bIdx  = group >> 1;
    const int h     = group & 1;

    const int tid   = threadIdx.x;
    const int lane  = tid & 31;
    const int wid   = tid >> 5;
    const int waveT = wid >> 1;             // 0..3
    const int waveK = wid & 1;              // 0..1

    const int tBase = h * HALF_T + tBlk * BT;   // global t of block row 0
    const int kBase = kBlk * BK;

    const float* Xbase = X + ((size_t)bIdx * T_DIM + tBase) * C_DIM;
    const float* Wbase = W + (size_t)kBase * C_DIM;

    // --- staging geometry: 64 rows x 8 quads (float4) = 512 slots, 2/thread
    const int s0row = tid >> 3,          s0q = (tid & 7);
    const int s1row = (tid + 256) >> 3,  s1q = (tid & 7);
    const unsigned gv0 = (unsigned)((s0row * C_DIM + s0q * 4) * 4);  // bytes
    const unsigned gv1 = (unsigned)((s1row * C_DIM + s1q * 4) * 4);

    const unsigned ldsX = (unsigned)(size_t)&Xs[0][0][0];  // LDS byte offsets
    const unsigned ldsW = (unsigned)(size_t)&Ws[0][0][0];
    unsigned xl0[2], xl1[2], wl0[2], wl1[2];
    #pragma unroll
    for (int bf = 0; bf < 2; ++bf) {
        const unsigned bo = (unsigned)(bf * BT * LDS_STR * 4);
        xl0[bf] = ldsX + bo + (unsigned)((s0row * LDS_STR + s0q * 4) * 4);
        xl1[bf] = ldsX + bo + (unsigned)((s1row * LDS_STR + s1q * 4) * 4);
        wl0[bf] = ldsW + bo + (unsigned)((s0row * LDS_STR + s0q * 4) * 4);
        wl1[bf] = ldsW + bo + (unsigned)((s1row * LDS_STR + s1q * 4) * 4);
    }

    v8f acc0 = {}, acc1 = {};

    const int m    = lane & 15;             // M (A) / N (B) index of this lane
    const int koff = (lane < 16) ? 0 : 2;   // K-pair owned by this lane half

    // Prefetch chunk 0 into buffer 0 (4 async b128 per thread per chunk).
    {
        const float* xp = Xbase;
        const float* wp = Wbase;
        ASYNC_B128(xl0[0], gv0, xp);
        ASYNC_B128(xl1[0], gv1, xp);
        ASYNC_B128(wl0[0], gv0, wp);
        ASYNC_B128(wl1[0], gv1, wp);
    }

    for (int i = 0; i < NCHUNK; ++i) {
        const int buf = i & 1;
        // Protect the buffer we are about to prefetch into: all waves must be
        // done computing chunk i-1 (which read buf^1).
        if (i > 0) __syncthreads();
        if (i + 1 < NCHUNK) {
            const float* xp = Xbase + (i + 1) * KC;
            const float* wp = Wbase + (i + 1) * KC;
            ASYNC_B128(xl0[buf ^ 1], gv0, xp);
            ASYNC_B128(xl1[buf ^ 1], gv1, xp);
            ASYNC_B128(wl0[buf ^ 1], gv0, wp);
            ASYNC_B128(wl1[buf ^ 1], gv1, wp);
            // Async loads complete in order: <=4 outstanding => chunk i landed.
            asm volatile("s_wait_asynccnt 0x4" ::: "memory");
        } else {
            asm volatile("s_wait_asynccnt 0x0" ::: "memory");
        }
        __syncthreads();   // everyone's chunk-i data visible in LDS

        // 8 WMMA K-steps of 4 cover the 32-wide chunk; A reused for 2 B tiles.
        // A 16x4 f32 layout: lanes 0-15 -> M, K={0,1}; lanes 16-31 -> K={2,3}.
        // B 4x16 symmetric with lane = N. float2 at col (kc+koff) matches.
        #pragma unroll
        for (int kk = 0; kk < 8; ++kk) {
            const int kc = kk * 4;
            v2f a  = *(const v2f*)&Xs[buf][waveT * 16 + m][kc + koff];
            v2f b0 = *(const v2f*)&Ws[buf][waveK * 32 + m][kc + koff];
            v2f b1 = *(const v2f*)&Ws[buf][waveK * 32 + 16 + m][kc + koff];
            acc0 = __builtin_amdgcn_wmma_f32_16x16x4_f32(
                false, a, false, b0, (short)0, acc0, false, false);
            acc1 = __builtin_amdgcn_wmma_f32_16x16x4_f32(
                false, a, false, b1, (short)0, acc1, false, false);
        }
    }

    // C/D 16x16 f32 layout: VGPR i, lanes 0-15 -> M=i, lanes 16-31 -> M=i+8,
    // N = lane%16. Add bias, lane-local argmax over 16 candidates, keeping
    // the earliest flat index (t asc, then k asc) on ties.
    const int kCol0 = kBase + waveK * 32 + m;
    const int kCol1 = kCol0 + 16;
    const float bv0 = bias[kCol0];
    const float bv1 = bias[kCol1];
    const int mBase = (lane < 16) ? 0 : 8;

    float bestV = acc0[0] + bv0;
    int bestC = 0;                       // candidate id = i*2 + j
    #pragma unroll
    for (int i = 0; i < 8; ++i) {
        #pragma unroll
        for (int j = 0; j < 2; ++j) {
            if (i == 0 && j == 0) continue;
            float v = (j ? acc1[i] + bv1 : acc0[i] + bv0);
            if (v > bestV) { bestV = v; bestC = i * 2 + j; }
        }
    }
    const int bi = bestC >> 1, bj = bestC & 1;
    const int tLocal = tBlk * BT + waveT * 16 + mBase + bi;   // in [0,1024)
    const int kWin   = (bj ? kCol1 : kCol0);
    const unsigned idx = (unsigned)(tLocal * K_OUT + kWin);   // flat column
    // High 32: monotone float key. Low 32: ~idx so equal values prefer the
    // SMALLER index under max (jnp.argmax first-occurrence semantics).
    unsigned long long key =
        ((unsigned long long)fmap(bestV) << 32) | (unsigned)(~idx);

    #pragma unroll
    for (int off = 16; off > 0; off >>= 1) {
        unsigned long long o = shflxor_u64(key, off);
        if (o > key) key = o;
    }
    if (lane == 0) wkeys[wid] = key;
    __syncthreads();
    if (tid == 0) {
        unsigned long long best = wkeys[0];
        #pragma unroll
        for (int i = 1; i < 8; ++i)
            if (wkeys[i] > best) best = wkeys[i];
        atomicMax(&keys[group], best);
    }
}

// ---------------------------------------------------------------------------
// Kernel 3: decode the 16 winners, scatter one 256-float codebook row each.
// hard one-hot value is exactly 1.0, so this is a pure gather-copy.
// ---------------------------------------------------------------------------
__global__ __launch_bounds__(256) void vq_scatter_kernel(
    const unsigned long long* __restrict__ keys,
    const float* __restrict__ codebook, float* __restrict__ out) {
    const int group = blockIdx.x;                       // 0..15
    const unsigned idx = ~(unsigned)(keys[group] & 0xffffffffull);
    const int tLocal = (int)(idx / K_OUT);
    const int k      = (int)(idx % K_OUT);
    const int g      = k / E_DIM;
    const int e      = k % E_DIM;
    const int bIdx   = group >> 1;
    const int h      = group & 1;
    const int t      = h * HALF_T + tLocal;

    const size_t outOff = ((size_t)(bIdx * T_DIM + t)) * (G_DIM * D_DIM)
                          + (size_t)g * D_DIM + threadIdx.x;
    const size_t cbOff  = ((size_t)(g * E_DIM + e)) * D_DIM + threadIdx.x;
    out[outOff] = codebook[cbOff];
}

// ---------------------------------------------------------------------------
extern "C" void kernel_launch(void* const* d_in, const int* in_sizes, int n_in,
                              void* d_out, int out_size, void* d_ws, size_t ws_size,
                              hipStream_t stream) {
    const float* X    = (const float*)d_in[0];   // (8,2048,1024)
    const float* W    = (const float*)d_in[1];   // (640,1024)
    const float* bias = (const float*)d_in[2];   // (640,)
    const float* cb   = (const float*)d_in[3];   // (640,256)
    float* out = (float*)d_out;                  // (8,2048,512)
    unsigned long long* keys = (unsigned long long*)d_ws;  // 16 slots

    const long n4 = (long)B_DIM * T_DIM * (G_DIM * D_DIM) / 4;  // float4 count
    vq_zero_kernel<<<1024, 256, 0, stream>>>((float4*)out, keys, n4);
    vq_gemm_argmax_kernel<<<N_GROUP * 16 * 10, 256, 0, stream>>>(X, W, bias, keys);
    vq_scatter_kernel<<<N_GROUP, 256, 0, stream>>>(keys, cb, out);
}